// HGT_90366111908556
// MI455X (gfx1250) — compile-verified
//
#include <hip/hip_runtime.h>
#include <math.h>

#define N_ENT 100000
#define N_USR 50000
#define N_EDGE 400000
#define N_INTER 500000
#define C 128
#define NHOPS 2

typedef __attribute__((ext_vector_type(2))) float v2f;
typedef __attribute__((ext_vector_type(8))) float v8f;
typedef __attribute__((ext_vector_type(4))) unsigned int v4u;
typedef __attribute__((ext_vector_type(8))) int v8i;
typedef __attribute__((ext_vector_type(4))) int v4i;

// ---------- helpers ----------
__device__ __forceinline__ unsigned f2ord(float f) {
    unsigned b = __float_as_uint(f);
    return (b & 0x80000000u) ? ~b : (b | 0x80000000u);
}
__device__ __forceinline__ float ord2f(unsigned u) {
    unsigned b = (u & 0x80000000u) ? (u & 0x7FFFFFFFu) : ~u;
    return __uint_as_float(b);
}

// ---------- init: copy embeddings into residual output (+ current buffer) ----------
__global__ void init_copy2(const float* __restrict__ src, float* __restrict__ d0,
                           float* __restrict__ d1, int n) {
    int i = blockIdx.x * blockDim.x + threadIdx.x;
    if (i < n) {
        float v = src[i];
        d0[i] = v;
        if (d1) d1[i] = v;
    }
}

// ---------- projection P = ent_cur @ W_Q via V_WMMA_F32_16X16X4_F32 ----------
// Block = 256 threads = 8 waves. Each wave owns one 16x16 output tile:
// block covers rows [16*bid, 16*bid+16), wave w covers cols [16w, 16w+16).
// W_Q (64KB, 128x128 f32) is staged into LDS by the Tensor Data Mover:
// one TENSOR_LOAD_TO_LDS DMA per workgroup (D# built per CDNA5 ISA
// §8.3/8.4), completion enforced with s_wait_tensorcnt + workgroup barrier.
__global__ __launch_bounds__(256) void project_wmma(const float* __restrict__ ent,
                                                    const float* __restrict__ Wq,
                                                    float* __restrict__ P) {
    __shared__ float lds_w[C * C];

#if __has_builtin(__builtin_amdgcn_tensor_load_to_lds)
    if (threadIdx.x == 0) {
        unsigned long long ga = (unsigned long long)(uintptr_t)Wq;
        // D# group 0 (128b): [1:0]=count=1 | [63:32]=lds_addr |
        //                    [120:64]=global_addr | [127:126]=type=2
        v4u g0;
        g0.x = 1u;
        g0.y = (unsigned)(uintptr_t)lds_w;                        // LDS byte offset
        g0.z = (unsigned)(ga & 0xFFFFFFFFu);                      // addr[31:0]
        g0.w = (unsigned)((ga >> 32) & 0x01FFFFFFu) | (2u << 30); // addr[56:32] | type=2
        // D# group 1 (256b): data_size=4B; 2D tensor 128x128 = tile 128x128,
        // tensor_dim0_stride = 128 elements.
        v8i g1;
        g1[0] = (int)(2u << 16);                 // wg_mask=0 | data_size=2 (4B)
        g1[1] = (int)((C & 0xFFFFu) << 16);      // tensor_dim0 lo16 (bits 63:48)
        g1[2] = (int)((C >> 16) | ((C & 0xFFFFu) << 16)); // dim0 hi16 | dim1 lo16
        g1[3] = (int)((C >> 16) | (C << 16));    // dim1 hi16 | tile_dim0=128
        g1[4] = (int)C;                          // tile_dim1=128 | tile_dim2=0
        g1[5] = (int)C;                          // tensor_dim0_stride lo32
        g1[6] = 0;                               // stride0 hi16 | dim1_stride lo16
        g1[7] = 0;                               // dim1_stride hi32
        v4i g2 = {0, 0, 0, 0};                   // <=2D: groups 2/3 unused
        v4i g3 = {0, 0, 0, 0};
        v8i g4 = {0, 0, 0, 0, 0, 0, 0, 0};       // 6-arg form: extra group, zero
        __builtin_amdgcn_tensor_load_to_lds(g0, g1, g2, g3, g4, 0);
        __builtin_amdgcn_s_wait_tensorcnt(0);
    }
    __syncthreads();
#else
    for (int i = threadIdx.x; i < C * C / 4; i += 256)
        ((float4*)lds_w)[i] = ((const float4*)Wq)[i];
    __syncthreads();
#endif

    const int wave = threadIdx.x >> 5;
    const int lane = threadIdx.x & 31;
    const int half = lane >> 4;      // 0: K={k0,k0+1}, 1: K={k0+2,k0+3}
    const int l16  = lane & 15;
    const int row0 = blockIdx.x * 16;
    const int n0   = wave * 16;
    const int koff = half * 2;

    const float* arow = ent + (size_t)(row0 + l16) * C;
    v8f acc = {};
#pragma unroll 4
    for (int k0 = 0; k0 < C; k0 += 4) {
        v2f a, b;
        // A 16x4: lanes 0-15 hold M=l16,K={k0,k0+1}; lanes 16-31 hold K={k0+2,k0+3}
        a.x = arow[k0 + koff];
        a.y = arow[k0 + koff + 1];
        // B 4x16: mirrored K split across lane halves, N = n0+l16
        b.x = lds_w[(k0 + koff) * C + n0 + l16];
        b.y = lds_w[(k0 + koff + 1) * C + n0 + l16];
        acc = __builtin_amdgcn_wmma_f32_16x16x4_f32(false, a, false, b,
                                                    (short)0, acc, false, false);
    }
    // D 16x16 f32: VGPR g -> row (g + 8*half), col n0+l16
    float* prow = P + (size_t)(row0 + half * 8) * C + n0 + l16;
#pragma unroll
    for (int g = 0; g < 8; ++g)
        prow[(size_t)g * C] = acc[g];
}

// ---------- pass 1: per-edge attention scores + segment max ----------
// One wave per edge; lanes 0-15 reduce head 0 (ch 0-63), lanes 16-31 head 1.
__global__ __launch_bounds__(256) void edge_score(const int* __restrict__ eidx,
                                                  const int* __restrict__ etype,
                                                  const float* __restrict__ P,
                                                  const float* __restrict__ rel_emb,
                                                  float* __restrict__ attn,
                                                  unsigned* __restrict__ seg_max) {
    int e = blockIdx.x * 8 + (threadIdx.x >> 5);
    if (e >= N_EDGE) return;
    int lane = threadIdx.x & 31;
    int head = eidx[e];
    int tail = eidx[N_EDGE + e];
    int rt = etype[e] - 1;
    float4 q = ((const float4*)(P + (size_t)head * C))[lane];
    float4 k = ((const float4*)(P + (size_t)tail * C))[lane];
    float4 r = ((const float4*)(rel_emb + (size_t)rt * C))[lane];
    float p = q.x * k.x * r.x + q.y * k.y * r.y + q.z * k.z * r.z + q.w * k.w * r.w;
#pragma unroll
    for (int off = 1; off < 16; off <<= 1)   // reduce within each 16-lane half
        p += __shfl_xor(p, off, 32);
    float score = p * 0.125f;                // / sqrt(DK=64)
    if ((lane & 15) == 0) {
        int h = lane >> 4;
        attn[(size_t)e * 2 + h] = score;
        atomicMax(&seg_max[(size_t)head * 2 + h], f2ord(score));
    }
}

// ---------- pass 2: exp(score - max) + segment sum ----------
__global__ void edge_exp(const int* __restrict__ eidx,
                         const unsigned* __restrict__ seg_max,
                         float* __restrict__ attn,
                         float* __restrict__ seg_sum) {
    int idx = blockIdx.x * blockDim.x + threadIdx.x;
    if (idx >= N_EDGE * 2) return;
    int e = idx >> 1, h = idx & 1;
    int head = eidx[e];
    float m = ord2f(seg_max[(size_t)head * 2 + h]);
    float ex = __expf(attn[idx] - m);
    attn[idx] = ex;
    atomicAdd(&seg_sum[(size_t)head * 2 + h], ex);
}

// ---------- pass 3: weighted message scatter-add into ent_agg ----------
__global__ __launch_bounds__(256) void edge_message(const int* __restrict__ eidx,
                                                    const int* __restrict__ etype,
                                                    const float* __restrict__ ent_cur,
                                                    const float* __restrict__ rel_emb,
                                                    const float* __restrict__ attn,
                                                    const float* __restrict__ seg_sum,
                                                    float* __restrict__ ent_agg) {
    int e = blockIdx.x * 8 + (threadIdx.x >> 5);
    if (e >= N_EDGE) return;
    int lane = threadIdx.x & 31;
    int head = eidx[e];
    int tail = eidx[N_EDGE + e];
    int rt = etype[e] - 1;
    int h = lane >> 4;
    float w = attn[(size_t)e * 2 + h] / seg_sum[(size_t)head * 2 + h];
    float4 v = ((const float4*)(ent_cur + (size_t)tail * C))[lane];
    float4 r = ((const float4*)(rel_emb + (size_t)rt * C))[lane];
    float* dst = ent_agg + (size_t)head * C + lane * 4;
    atomicAdd(dst + 0, v.x * r.x * w);
    atomicAdd(dst + 1, v.y * r.y * w);
    atomicAdd(dst + 2, v.z * r.z * w);
    atomicAdd(dst + 3, v.w * r.w * w);
}

// ---------- sparse user aggregation: usr_agg[r] += val * ent_cur[c] ----------
__global__ __launch_bounds__(256) void user_agg_k(const int* __restrict__ rows,
                                                  const int* __restrict__ cols,
                                                  const float* __restrict__ vals,
                                                  const float* __restrict__ ent_cur,
                                                  float* __restrict__ usr_agg) {
    int i = blockIdx.x * 8 + (threadIdx.x >> 5);
    if (i >= N_INTER) return;
    int lane = threadIdx.x & 31;
    int r = rows[i], c = cols[i];
    float v = vals[i];
    float4 e4 = ((const float4*)(ent_cur + (size_t)c * C))[lane];
    float* dst = usr_agg + (size_t)r * C + lane * 4;
    atomicAdd(dst + 0, v * e4.x);
    atomicAdd(dst + 1, v * e4.y);
    atomicAdd(dst + 2, v * e4.z);
    atomicAdd(dst + 3, v * e4.w);
}

// ---------- L2-normalize row, write next-hop buffer, accumulate residual ----------
__global__ __launch_bounds__(256) void finalize_norm(const float* __restrict__ agg,
                                                     float* __restrict__ cur_out,
                                                     float* __restrict__ res, int nrows) {
    int row = blockIdx.x * 8 + (threadIdx.x >> 5);
    if (row >= nrows) return;
    int lane = threadIdx.x & 31;
    float4 a = ((const float4*)(agg + (size_t)row * C))[lane];
    float ss = a.x * a.x + a.y * a.y + a.z * a.z + a.w * a.w;
#pragma unroll
    for (int off = 1; off < 32; off <<= 1)
        ss += __shfl_xor(ss, off, 32);
    float inv = 1.0f / fmaxf(sqrtf(ss), 1e-12f);
    float4 o = {a.x * inv, a.y * inv, a.z * inv, a.w * inv};
    if (cur_out) ((float4*)(cur_out + (size_t)row * C))[lane] = o;
    float* rp = res + (size_t)row * C + lane * 4;
    rp[0] += o.x; rp[1] += o.y; rp[2] += o.z; rp[3] += o.w;
}

extern "C" void kernel_launch(void* const* d_in, const int* in_sizes, int n_in,
                              void* d_out, int out_size, void* d_ws, size_t ws_size,
                              hipStream_t stream) {
    const float* user_emb   = (const float*)d_in[0];
    const float* entity_emb = (const float*)d_in[1];
    const int*   edge_index = (const int*)d_in[2];
    const int*   edge_type  = (const int*)d_in[3];
    const int*   inter_rows = (const int*)d_in[4];
    const int*   inter_cols = (const int*)d_in[5];
    const float* inter_vals = (const float*)d_in[6];
    const float* W_Q        = (const float*)d_in[7];
    const float* rel_emb    = (const float*)d_in[8];

    float* out_ent = (float*)d_out;
    float* out_usr = out_ent + (size_t)N_ENT * C;

    // workspace layout (P is reused as ent_agg after the exp pass)
    float*    P       = (float*)d_ws;                       // [N_ENT, C]
    float*    ent_cur = P + (size_t)N_ENT * C;              // [N_ENT, C]
    float*    usr_agg = ent_cur + (size_t)N_ENT * C;        // [N_USR, C]
    float*    attn    = usr_agg + (size_t)N_USR * C;        // [N_EDGE, 2]
    unsigned* seg_max = (unsigned*)(attn + (size_t)N_EDGE * 2); // [N_ENT, 2]
    float*    seg_sum = (float*)(seg_max + (size_t)N_ENT * 2);  // [N_ENT, 2]

    {
        int n = N_ENT * C;
        init_copy2<<<(n + 255) / 256, 256, 0, stream>>>(entity_emb, out_ent, ent_cur, n);
        n = N_USR * C;
        init_copy2<<<(n + 255) / 256, 256, 0, stream>>>(user_emb, out_usr, nullptr, n);
    }

    for (int hop = 0; hop < NHOPS; ++hop) {
        (void)hipMemsetAsync(seg_max, 0, sizeof(unsigned) * (size_t)N_ENT * 2, stream); // ordered-0 == -inf
        (void)hipMemsetAsync(seg_sum, 0, sizeof(float) * (size_t)N_ENT * 2, stream);
        (void)hipMemsetAsync(usr_agg, 0, sizeof(float) * (size_t)N_USR * C, stream);

        project_wmma<<<N_ENT / 16, 256, 0, stream>>>(ent_cur, W_Q, P);
        edge_score<<<(N_EDGE + 7) / 8, 256, 0, stream>>>(edge_index, edge_type, P,
                                                         rel_emb, attn, seg_max);
        edge_exp<<<(N_EDGE * 2 + 255) / 256, 256, 0, stream>>>(edge_index, seg_max,
                                                               attn, seg_sum);
        (void)hipMemsetAsync(P, 0, sizeof(float) * (size_t)N_ENT * C, stream); // P -> ent_agg
        edge_message<<<(N_EDGE + 7) / 8, 256, 0, stream>>>(edge_index, edge_type, ent_cur,
                                                           rel_emb, attn, seg_sum, P);
        user_agg_k<<<(N_INTER + 7) / 8, 256, 0, stream>>>(inter_rows, inter_cols,
                                                          inter_vals, ent_cur, usr_agg);
        finalize_norm<<<(N_ENT + 7) / 8, 256, 0, stream>>>(P, ent_cur, out_ent, N_ENT);
        finalize_norm<<<(N_USR + 7) / 8, 256, 0, stream>>>(usr_agg, nullptr, out_usr, N_USR);
    }
}